// Ellp2Mask_10557029613720
// MI455X (gfx1250) — compile-verified
//
#include <hip/hip_runtime.h>
#include <math.h>

// Ellipse conic -> soft mask over a 1024x1024 grid.
// dist[e,p] = [A,Bc,C,D,E,F]_e . [gx^2, gx*gy, gy^2, gx, gy, 1]_p
// Implemented as two chained V_WMMA_F32_16X16X4_F32 (K=0..3, K=4..7 zero-padded),
// with -EFFECTIVE_SCALE folded into the coefficients so the accumulator is the logit.
// Store-bandwidth bound (256 MB out, ~11us at 23.3 TB/s); non-temporal stores since
// the output exceeds the 192MB L2 and is never re-read. Sigmoid = v_exp_f32 + v_rcp_f32.

typedef __attribute__((ext_vector_type(2))) float v2f;
typedef __attribute__((ext_vector_type(8))) float v8f;

#define HH 1024
#define WW 1024
#define HWPIX (HH * WW)
#define NB 32
#define EPSV 1e-8f
#define EFF_SCALE 10.0f

__global__ __launch_bounds__(256) void ellp2mask_wmma(
    const float* __restrict__ params,   // (32,5): x0,y0,a,b,theta
    float* __restrict__ out_mask,       // (32,1,1024,1024)
    float* __restrict__ out_logits)     // (32,1,1024,1024)
{
    const int lane = threadIdx.x & 31;
    const int wave = (blockIdx.x << 3) + (threadIdx.x >> 5);  // 65536 waves
    const int half = lane >> 4;     // which half of the wave
    const int sub  = lane & 15;     // N (pixel) / M (ellipse) sub-index

    // ---- B fragment: pixel features for column N = sub of this tile ----
    // Layout (consistent with documented 32-bit A 16x4 layout):
    //   v0: lanes0-15 = row K0, lanes16-31 = row K2 ; v1: K1 / K3
    const int p  = wave * 16 + sub;             // flat pixel index
    const float gx = (float)(p & (WW - 1));     // column j
    const float gy = (float)(p >> 10);          // row i
    v2f bf0, bf1;
    bf0.x = half ? gy * gy : gx * gx;           // K2 : K0
    bf0.y = half ? gx      : gx * gy;           // K3 : K1
    bf1.x = half ? 0.0f    : gy;                // K6 : K4
    bf1.y = half ? 0.0f    : 1.0f;              // K7 : K5

#pragma unroll
    for (int tile = 0; tile < 2; ++tile) {      // ellipse M-tiles: 0-15, 16-31
        // ---- A fragment: conic coefficients for ellipse M = tile*16 + sub ----
        const int e = tile * 16 + sub;
        const float x0 = params[e * 5 + 0];
        const float y0 = params[e * 5 + 1];
        const float ea = params[e * 5 + 2];
        const float eb = params[e * 5 + 3];
        const float th = params[e * 5 + 4];
        float sn, cs;
        __sincosf(th, &sn, &cs);
        const float a2i = 1.0f / (ea * ea + EPSV);
        const float b2i = 1.0f / (eb * eb + EPSV);
        const float cA = sn * sn * b2i + cs * cs * a2i;
        const float cB = 2.0f * (a2i - b2i) * sn * cs;
        const float cC = cs * cs * b2i + sn * sn * a2i;
        const float cD = -2.0f * cA * x0 - cB * y0;
        const float cE = -cB * x0 - 2.0f * cC * y0;
        const float cF = -(cD * x0 + cE * y0) * 0.5f - 1.0f;
        const float k = -EFF_SCALE;             // fold logits scale into coeffs
        v2f af0, af1;
        af0.x = k * (half ? cC : cA);           // K2 : K0
        af0.y = k * (half ? cD : cB);           // K3 : K1
        af1.x = half ? 0.0f : k * cE;           // K6 : K4
        af1.y = half ? 0.0f : k * cF;           // K7 : K5

        v8f acc = {};
        acc = __builtin_amdgcn_wmma_f32_16x16x4_f32(
            false, af0, false, bf0, (short)0, acc, false, false);
        acc = __builtin_amdgcn_wmma_f32_16x16x4_f32(
            false, af1, false, bf1, (short)0, acc, false, false);

        // ---- D fragment: acc[v] = logit for (ellipse tile*16+v(+8), pixel sub) ----
#pragma unroll
        for (int v = 0; v < 8; ++v) {
            const int erow = tile * 16 + v + (half ? 8 : 0);
            const long idx = (long)erow * HWPIX + (long)(wave * 16 + sub);
            const float logit = acc[v];
            // sigmoid via hardware reciprocal: v_exp_f32 + v_add_f32 + v_rcp_f32
            const float m = __builtin_amdgcn_rcpf(1.0f + __expf(-logit));
            __builtin_nontemporal_store(m, out_mask + idx);
            __builtin_nontemporal_store(logit, out_logits + idx);
        }
    }
}

extern "C" void kernel_launch(void* const* d_in, const int* in_sizes, int n_in,
                              void* d_out, int out_size, void* d_ws, size_t ws_size,
                              hipStream_t stream) {
    const float* params = (const float*)d_in[0];       // 32*5 floats
    float* out = (float*)d_out;                        // mask then logits
    float* out_mask = out;
    float* out_logits = out + (long)NB * HWPIX;

    // 65536 pixel tiles of 16 px; 8 waves (256 threads) per block -> 8192 blocks
    const int blocks = HWPIX / 16 / 8;
    ellp2mask_wmma<<<blocks, 256, 0, stream>>>(params, out_mask, out_logits);
}